// Memory_48017734369831
// MI455X (gfx1250) — compile-verified
//
#include <hip/hip_runtime.h>
#include <hip/hip_bf16.h>
#include <math.h>

// ---------------------------------------------------------------------------
// Shapes: B=32, N=256, D=2048, d=512, M=512  ->  rows R = B*N = 8192
// Outputs (flat): p_score_nor[8192] | updated_p_score[8192] |
//                 nor_loss[8192] | abn_loss[8192] | ref_nor_e[8192*512]
// ---------------------------------------------------------------------------

typedef __attribute__((ext_vector_type(2))) float v2f;
typedef __attribute__((ext_vector_type(8))) float v8f;

__device__ __forceinline__ v8f wmma_f32_16x16x4(v2f a, v2f b, v8f c) {
  // emits v_wmma_f32_16x16x4_f32
  return __builtin_amdgcn_wmma_f32_16x16x4_f32(
      /*neg_a=*/false, a, /*neg_b=*/false, b,
      /*c_mod=*/(short)0, c, /*reuse_a=*/false, /*reuse_b=*/false);
}

// ---------------------------------------------------------------------------
// Kernel 1: E = relu(conv1d(x, w, SAME) + bias), as GEMM M=8192 N=512 K=3*2048
//   x: (32,256,2048)  w: (512,2048,3)  bias: (512)  E: (8192,512)
//   Block tile 128x128, 8 waves in 4x2 grid, each wave 32x64 (8 accumulators).
//   B staged as K-pairs (v2f) -> B fragment = one ds_load_b64, no repack movs.
//   Double-buffered LDS, one barrier per 32-wide K chunk, 192 chunks.
// ---------------------------------------------------------------------------
__global__ __launch_bounds__(256) void conv_embed_wmma(
    const float* __restrict__ x, const float* __restrict__ w,
    const float* __restrict__ bias, float* __restrict__ E) {
  __shared__ float As[2][128][33];   // M x K, padded (gcd(33,64)=1)
  __shared__ v2f   Bp[2][16][130];   // K-pair x N (128 + 2 pad), 8B elements

  const int tid  = threadIdx.x;
  const int bm   = blockIdx.x * 128;         // output-row base
  const int bn   = blockIdx.y * 128;         // output-col base
  const int lane = tid & 31;
  const int wv   = tid >> 5;
  const int wm   = (wv >> 1) << 5;           // 0,32,64,96
  const int wn   = (wv & 1) << 6;            // 0,64
  const int half = lane >> 4;
  const int lr   = lane & 15;

  const int bb    = bm >> 8;                 // batch (rows never cross batch)
  const int nbase = bm & 255;

  v8f acc[2][4];
  #pragma unroll
  for (int i = 0; i < 2; ++i)
    #pragma unroll
    for (int j = 0; j < 4; ++j) acc[i][j] = (v8f){};

  float4 ra[4];
  float  rb[16];
  const int bnn = tid & 127;                 // B staging column
  const int bkg = (tid >> 7) << 4;           // 0 or 16 (K group)

  auto load_chunk = [&](int c) {
    const int tap = c >> 6;                  // 0..2
    const int kb  = (c & 63) << 5;           // 0..2016
    #pragma unroll
    for (int r = 0; r < 4; ++r) {
      int idx  = tid + (r << 8);
      int row  = idx >> 3;                   // 0..127
      int seg  = (idx & 7) << 2;             // 0..28
      int n_in = nbase + row + tap - 1;
      float4 v = make_float4(0.f, 0.f, 0.f, 0.f);
      if ((unsigned)n_in < 256u) {
        v = *reinterpret_cast<const float4*>(
            x + (size_t)(bb * 256 + n_in) * 2048 + kb + seg);
      }
      ra[r] = v;
    }
    const float* wp = w + (size_t)(bn + bnn) * (2048 * 3)
                        + (size_t)(kb + bkg) * 3 + tap;
    #pragma unroll
    for (int k = 0; k < 16; ++k) rb[k] = wp[k * 3];
  };
  auto store_chunk = [&](int buf) {
    #pragma unroll
    for (int r = 0; r < 4; ++r) {
      int idx = tid + (r << 8);
      int row = idx >> 3;
      int seg = (idx & 7) << 2;
      As[buf][row][seg + 0] = ra[r].x; As[buf][row][seg + 1] = ra[r].y;
      As[buf][row][seg + 2] = ra[r].z; As[buf][row][seg + 3] = ra[r].w;
    }
    #pragma unroll
    for (int i = 0; i < 8; ++i) {
      v2f p; p.x = rb[2 * i]; p.y = rb[2 * i + 1];
      Bp[buf][(bkg >> 1) + i][bnn] = p;
    }
  };

  const int NC = 192;                        // 3 taps * 64 chunks
  load_chunk(0);
  store_chunk(0);
  __syncthreads();
  for (int c = 0; c < NC; ++c) {
    const int buf = c & 1;
    if (c + 1 < NC) load_chunk(c + 1);       // prefetch next chunk into regs
    #pragma unroll
    for (int kk = 0; kk < 32; kk += 4) {
      const int kp = (kk >> 1) + half;       // K-pair row for this half-wave
      v2f a0, a1;
      a0.x = As[buf][wm + lr][kk + 2 * half + 0];
      a0.y = As[buf][wm + lr][kk + 2 * half + 1];
      a1.x = As[buf][wm + 16 + lr][kk + 2 * half + 0];
      a1.y = As[buf][wm + 16 + lr][kk + 2 * half + 1];
      #pragma unroll
      for (int j = 0; j < 4; ++j) {
        v2f b = Bp[buf][kp][wn + j * 16 + lr];
        acc[0][j] = wmma_f32_16x16x4(a0, b, acc[0][j]);
        acc[1][j] = wmma_f32_16x16x4(a1, b, acc[1][j]);
      }
    }
    if (c + 1 < NC) store_chunk(buf ^ 1);    // fill the other buffer
    __syncthreads();
  }

  // ---- epilogue: bias + relu, D layout: VGPR v -> M = v + 8*half ----
  #pragma unroll
  for (int j = 0; j < 4; ++j) {
    const int col = bn + wn + j * 16 + lr;
    const float bz = bias[col];
    #pragma unroll
    for (int v = 0; v < 8; ++v) {
      int row0 = bm + wm + half * 8 + v;
      int row1 = row0 + 16;
      float r0 = acc[0][j][v] + bz; r0 = r0 > 0.f ? r0 : 0.f;
      float r1 = acc[1][j][v] + bz; r1 = r1 > 0.f ? r1 : 0.f;
      E[(size_t)row0 * 512 + col] = r0;
      E[(size_t)row1 * 512 + col] = r1;
    }
  }
}

// ---------------------------------------------------------------------------
// Kernel 2: S = E @ keys^T   (M=8192, N=1024, K=512)
//   keys rows 0..511 = nor_keys, 512..1023 = abn_keys (selected per block;
//   128-col blocks never straddle the 512 boundary).
//   Same 128x128 / 2x4-wave-tile / pair-layout / double-buffered structure.
// ---------------------------------------------------------------------------
__global__ __launch_bounds__(256) void score_wmma(
    const float* __restrict__ E, const float* __restrict__ nor_keys,
    const float* __restrict__ abn_keys, float* __restrict__ S) {
  __shared__ float As[2][128][33];
  __shared__ v2f   Bp[2][16][130];

  const int tid  = threadIdx.x;
  const int bm   = blockIdx.x * 128;
  const int bn   = blockIdx.y * 128;         // 0..896
  const int lane = tid & 31;
  const int wv   = tid >> 5;
  const int wm   = (wv >> 1) << 5;
  const int wn   = (wv & 1) << 6;
  const int half = lane >> 4;
  const int lr   = lane & 15;

  const float* kptr = (bn < 512) ? (nor_keys + (size_t)bn * 512)
                                 : (abn_keys + (size_t)(bn - 512) * 512);

  v8f acc[2][4];
  #pragma unroll
  for (int i = 0; i < 2; ++i)
    #pragma unroll
    for (int j = 0; j < 4; ++j) acc[i][j] = (v8f){};

  float4 ra[4];
  float4 rb[4];
  const int bnn = tid & 127;
  const int bkg = (tid >> 7) << 4;           // 0 or 16

  auto load_chunk = [&](int c) {
    const int kb = c << 5;
    #pragma unroll
    for (int r = 0; r < 4; ++r) {
      int idx = tid + (r << 8);
      int row = idx >> 3;
      int seg = (idx & 7) << 2;
      ra[r] = *reinterpret_cast<const float4*>(
          E + (size_t)(bm + row) * 512 + kb + seg);
    }
    const float* kp = kptr + (size_t)bnn * 512 + kb + bkg;
    #pragma unroll
    for (int q = 0; q < 4; ++q)
      rb[q] = *reinterpret_cast<const float4*>(kp + 4 * q);
  };
  auto store_chunk = [&](int buf) {
    #pragma unroll
    for (int r = 0; r < 4; ++r) {
      int idx = tid + (r << 8);
      int row = idx >> 3;
      int seg = (idx & 7) << 2;
      As[buf][row][seg + 0] = ra[r].x; As[buf][row][seg + 1] = ra[r].y;
      As[buf][row][seg + 2] = ra[r].z; As[buf][row][seg + 3] = ra[r].w;
    }
    #pragma unroll
    for (int q = 0; q < 4; ++q) {
      v2f p0; p0.x = rb[q].x; p0.y = rb[q].y;
      v2f p1; p1.x = rb[q].z; p1.y = rb[q].w;
      Bp[buf][(bkg >> 1) + 2 * q + 0][bnn] = p0;
      Bp[buf][(bkg >> 1) + 2 * q + 1][bnn] = p1;
    }
  };

  const int NC = 16;                         // 512 / 32
  load_chunk(0);
  store_chunk(0);
  __syncthreads();
  for (int c = 0; c < NC; ++c) {
    const int buf = c & 1;
    if (c + 1 < NC) load_chunk(c + 1);
    #pragma unroll
    for (int kk = 0; kk < 32; kk += 4) {
      const int kp = (kk >> 1) + half;
      v2f a0, a1;
      a0.x = As[buf][wm + lr][kk + 2 * half + 0];
      a0.y = As[buf][wm + lr][kk + 2 * half + 1];
      a1.x = As[buf][wm + 16 + lr][kk + 2 * half + 0];
      a1.y = As[buf][wm + 16 + lr][kk + 2 * half + 1];
      #pragma unroll
      for (int j = 0; j < 4; ++j) {
        v2f b = Bp[buf][kp][wn + j * 16 + lr];
        acc[0][j] = wmma_f32_16x16x4(a0, b, acc[0][j]);
        acc[1][j] = wmma_f32_16x16x4(a1, b, acc[1][j]);
      }
    }
    if (c + 1 < NC) store_chunk(buf ^ 1);
    __syncthreads();
  }

  #pragma unroll
  for (int j = 0; j < 4; ++j) {
    const int col = bn + wn + j * 16 + lr;
    #pragma unroll
    for (int v = 0; v < 8; ++v) {
      int row0 = bm + wm + half * 8 + v;
      int row1 = row0 + 16;
      S[(size_t)row0 * 1024 + col] = acc[0][j][v];
      S[(size_t)row1 * 1024 + col] = acc[1][j][v];
    }
  }
}

// ---------------------------------------------------------------------------
// Kernel 3: t[j] = sum_k keys[j][k] * cls_mem_w[512+k]   (j = 0..1023)
// ---------------------------------------------------------------------------
__global__ __launch_bounds__(128) void keys_dot_w2(
    const float* __restrict__ nor_keys, const float* __restrict__ abn_keys,
    const float* __restrict__ cls_mem_w, float* __restrict__ t) {
  __shared__ float red[128];
  const int j = blockIdx.x, tid = threadIdx.x;
  const float* kp = (j < 512) ? (nor_keys + (size_t)j * 512)
                              : (abn_keys + (size_t)(j - 512) * 512);
  float s = 0.f;
  for (int k = tid; k < 512; k += 128) s += kp[k] * cls_mem_w[512 + k];
  red[tid] = s; __syncthreads();
  for (int off = 64; off > 0; off >>= 1) {
    if (tid < off) red[tid] += red[tid + off];
    __syncthreads();
  }
  if (tid == 0) t[j] = red[0];
}

// ---------------------------------------------------------------------------
// Kernel 4: per-row epilogue (one block per row, 256 threads, wave32-safe)
// ---------------------------------------------------------------------------
__device__ __forceinline__ float block_sum(float v, float* red, int tid) {
  __syncthreads();
  red[tid] = v; __syncthreads();
  for (int off = 128; off > 0; off >>= 1) {
    if (tid < off) red[tid] += red[tid + off];
    __syncthreads();
  }
  return red[0];
}
__device__ __forceinline__ float block_max(float v, float* red, int tid) {
  __syncthreads();
  red[tid] = v; __syncthreads();
  for (int off = 128; off > 0; off >>= 1) {
    if (tid < off) red[tid] = fmaxf(red[tid], red[tid + off]);
    __syncthreads();
  }
  return red[0];
}
__device__ __forceinline__ int block_argmax(float v, int idx, float* red,
                                            int* redi, int tid) {
  __syncthreads();
  red[tid] = v; redi[tid] = idx; __syncthreads();
  for (int off = 128; off > 0; off >>= 1) {
    if (tid < off && red[tid + off] > red[tid]) {
      red[tid] = red[tid + off]; redi[tid] = redi[tid + off];
    }
    __syncthreads();
  }
  return redi[0];
}

__global__ __launch_bounds__(256) void row_epilogue(
    const float* __restrict__ S, const float* __restrict__ E,
    const float* __restrict__ nor_keys, const float* __restrict__ abn_keys,
    const float* __restrict__ cls_w, const float* __restrict__ cls_b,
    const float* __restrict__ cls_mem_w, const float* __restrict__ cls_mem_b,
    const float* __restrict__ t, float* __restrict__ out_p,
    float* __restrict__ out_up, float* __restrict__ out_ln,
    float* __restrict__ out_la) {
  __shared__ float red[256];
  __shared__ int redi[256];
  const int r = blockIdx.x, tid = threadIdx.x;
  const float* Sr = S + (size_t)r * 1024;
  const float* Er = E + (size_t)r * 512;

  const float s0 = Sr[tid], s1 = Sr[tid + 256];
  const float s2 = Sr[tid + 512], s3 = Sr[tid + 768];

  // softmax over all 1024 columns
  const float m =
      block_max(fmaxf(fmaxf(s0, s1), fmaxf(s2, s3)), red, tid);
  const float e0 = expf(s0 - m), e1 = expf(s1 - m);
  const float e2 = expf(s2 - m), e3 = expf(s3 - m);
  const float sumexp = block_sum(e0 + e1 + e2 + e3, red, tid);
  const float pt = block_sum(
      e0 * t[tid] + e1 * t[tid + 256] + e2 * t[tid + 512] + e3 * t[tid + 768],
      red, tid);
  const float mem_term = pt / sumexp;   // = (softmax @ keys) . cls_mem_w[512:]

  // per-half argmax (nor: cols 0..511, abn: cols 512..1023)
  float hv0 = s0; int hi0 = tid;
  if (s1 > hv0) { hv0 = s1; hi0 = tid + 256; }
  const int idx0 = block_argmax(hv0, hi0, red, redi, tid);
  float hv1 = s2; int hi1 = tid + 512;
  if (s3 > hv1) { hv1 = s3; hi1 = tid + 768; }
  const int idx1 = block_argmax(hv1, hi1, red, redi, tid);

  // dot products against E row
  const float ea = Er[tid], eb = Er[tid + 256];
  const float pcls = block_sum(ea * cls_w[tid] + eb * cls_w[tid + 256], red, tid);
  const float pw1 =
      block_sum(ea * cls_mem_w[tid] + eb * cls_mem_w[tid + 256], red, tid);

  // compactness losses vs nearest key of each bank
  const float* kn = nor_keys + (size_t)idx0 * 512;
  float d0 = ea - kn[tid], d1 = eb - kn[tid + 256];
  const float ln = block_sum(d0 * d0 + d1 * d1, red, tid);
  const float* ka = abn_keys + (size_t)(idx1 - 512) * 512;
  d0 = ea - ka[tid]; d1 = eb - ka[tid + 256];
  const float la = block_sum(d0 * d0 + d1 * d1, red, tid);

  if (tid == 0) {
    out_p[r]  = 1.f / (1.f + expf(-(pcls + cls_b[0])));
    out_up[r] = 1.f / (1.f + expf(-(pw1 + mem_term + cls_mem_b[0])));
    out_ln[r] = ln * (1.f / 512.f);
    out_la[r] = la * (1.f / 512.f);
  }
}

// ---------------------------------------------------------------------------
extern "C" void kernel_launch(void* const* d_in, const int* in_sizes, int n_in,
                              void* d_out, int out_size, void* d_ws,
                              size_t ws_size, hipStream_t stream) {
  const float* ref_nor   = (const float*)d_in[0];
  // d_in[1] = ref_abn : dead code (only feeds discarded p_score[B:])
  const float* nor_keys  = (const float*)d_in[2];
  const float* abn_keys  = (const float*)d_in[3];
  const float* conv_w    = (const float*)d_in[4];
  const float* conv_b    = (const float*)d_in[5];
  const float* cls_w     = (const float*)d_in[6];
  const float* cls_b     = (const float*)d_in[7];
  const float* cls_mem_w = (const float*)d_in[8];
  const float* cls_mem_b = (const float*)d_in[9];

  float* out = (float*)d_out;
  float* out_p  = out;                 // p_score_nor        [8192]
  float* out_up = out + 8192;          // updated_p_score    [8192]
  float* out_ln = out + 16384;         // nor_compactness    [8192]
  float* out_la = out + 24576;         // abn_compactness    [8192]
  float* E      = out + 32768;         // ref_nor_e          [8192*512]

  float* S = (float*)d_ws;             // scores             [8192*1024]
  float* t = S + (size_t)8192 * 1024;  // keys @ cls_mem_w[512:] [1024]

  // 1) E = relu(conv(x) + b)  — WMMA GEMM, M=8192 N=512 K=6144
  conv_embed_wmma<<<dim3(64, 4), 256, 0, stream>>>(ref_nor, conv_w, conv_b, E);
  // 2) S = E @ keys^T        — WMMA GEMM, M=8192 N=1024 K=512
  score_wmma<<<dim3(64, 8), 256, 0, stream>>>(E, nor_keys, abn_keys, S);
  // 3) t = keys @ cls_mem_w[512:]
  keys_dot_w2<<<1024, 128, 0, stream>>>(nor_keys, abn_keys, cls_mem_w, t);
  // 4) per-row softmax / argmax / losses / sigmoids
  row_epilogue<<<8192, 256, 0, stream>>>(S, E, nor_keys, abn_keys, cls_w, cls_b,
                                         cls_mem_w, cls_mem_b, t, out_p, out_up,
                                         out_ln, out_la);
}